// GCNEncoder_13142599925967
// MI455X (gfx1250) — compile-verified
//
#include <hip/hip_runtime.h>

typedef __attribute__((ext_vector_type(2))) float v2f;
typedef __attribute__((ext_vector_type(8))) float v8f;

// ---------------------------------------------------------------------------
// 1) Degree / norm precompute
// ---------------------------------------------------------------------------
__global__ void k_deg_init(float* __restrict__ deg, int N) {
    int i = blockIdx.x * blockDim.x + threadIdx.x;
    if (i < N) deg[i] = 1.0f;                       // self-loop weight
}

__global__ void k_deg_scatter(const long long* __restrict__ ei,
                              const float* __restrict__ w,
                              float* __restrict__ deg, int E) {
    int e = blockIdx.x * blockDim.x + threadIdx.x;
    if (e < E) {
        int d = (int)ei[(size_t)E + e];             // dst row of edge_index
        atomicAdd(&deg[d], w[e]);
    }
}

__global__ void k_dinv(float* __restrict__ deg, int N) {
    int i = blockIdx.x * blockDim.x + threadIdx.x;
    if (i < N) {
        float d = deg[i];
        deg[i] = (d > 0.0f) ? rsqrtf(d) : 0.0f;     // in-place -> dinv
    }
}

__global__ void k_norm(const long long* __restrict__ ei,
                       const float* __restrict__ w,
                       const float* __restrict__ dinv,
                       float* __restrict__ norm,
                       int* __restrict__ src32, int* __restrict__ dst32, int E) {
    int e = blockIdx.x * blockDim.x + threadIdx.x;
    if (e < E) {
        int s = (int)ei[e];
        int d = (int)ei[(size_t)E + e];
        src32[e] = s;
        dst32[e] = d;
        norm[e] = dinv[s] * w[e] * dinv[d];
    }
}

// ---------------------------------------------------------------------------
// 2) Dense transform h = X @ W via V_WMMA_F32_16X16X4_F32.
//    One wave per 16x16 output tile; 8 waves / block (wave32).
//    A 16x4 layout : lanes 0-15 -> K=k0+0,+1 ; lanes 16-31 -> K=k0+2,+3
//    B 4x16 layout : same K split, N striped over lanes within a VGPR
//    C/D 16x16     : VGPR v -> row tm+v (lanes 0-15) / tm+v+8 (lanes 16-31)
// ---------------------------------------------------------------------------
__global__ __launch_bounds__(256) void
k_gemm_wmma(const float* __restrict__ X, const float* __restrict__ W,
            float* __restrict__ H, int N, int din, int dout, int totalTiles) {
    int wave = threadIdx.x >> 5;
    int lane = threadIdx.x & 31;
    int tile = blockIdx.x * 8 + wave;
    if (tile >= totalTiles) return;                 // wave-uniform branch

    int tilesN = dout >> 4;
    int tm = (tile / tilesN) << 4;
    int tn = (tile % tilesN) << 4;

    int half = lane >> 4;                           // 0: lanes 0-15, 1: 16-31
    int l16  = lane & 15;

    int arow = tm + l16;
    if (arow >= N) arow = N - 1;                    // clamp (keep EXEC all-1)

    const float* ap = X + (size_t)arow * din + 2 * half;
    const float* bp = W + (size_t)(2 * half) * dout + tn + l16;

    v8f acc = {};
#pragma unroll 4
    for (int k0 = 0; k0 < din; k0 += 4) {
        v2f a; a.x = ap[0];    a.y = ap[1];         // contiguous -> b64 load
        v2f b; b.x = bp[0];    b.y = bp[dout];
        acc = __builtin_amdgcn_wmma_f32_16x16x4_f32(
            false, a, false, b, (short)0, acc, false, false);
        ap += 4;
        bp += (size_t)4 * dout;
    }

#pragma unroll
    for (int v = 0; v < 8; ++v) {
        int r = tm + v + 8 * half;
        if (r < N) H[(size_t)r * dout + tn + l16] = acc[v];
    }
}

// ---------------------------------------------------------------------------
// 3) Aggregation: out = h*dinv^2 (self-loop, full overwrite) then edge
//    scatter-add, then bias (+ optional ReLU).
// ---------------------------------------------------------------------------
__global__ void k_selfloop(const float* __restrict__ H,
                           const float* __restrict__ dinv,
                           float* __restrict__ Y, int N, int dout) {
    int chunks = dout >> 2;                          // float4 per thread
    long long t = (long long)blockIdx.x * blockDim.x + threadIdx.x;
    if (t >= (long long)N * chunks) return;
    int i = (int)(t / chunks);
    int c = (int)(t % chunks);
    float di = dinv[i];
    float s = di * di;
    float4 h = ((const float4*)(H + (size_t)i * dout))[c];
    float4 o; o.x = h.x * s; o.y = h.y * s; o.z = h.z * s; o.w = h.w * s;
    ((float4*)(Y + (size_t)i * dout))[c] = o;
}

__global__ void k_scatter(const float* __restrict__ H,
                          const int* __restrict__ src,
                          const int* __restrict__ dst,
                          const float* __restrict__ norm,
                          float* __restrict__ Y, int E, int dout) {
    int chunks = dout >> 2;
    long long t = (long long)blockIdx.x * blockDim.x + threadIdx.x;
    if (t >= (long long)E * chunks) return;
    int e = (int)(t / chunks);                       // one wave covers 1 edge
    int c = (int)(t % chunks);                       //   when dout == 128
    int s = src[e];
    int d = dst[e];
    float n = norm[e];
    float4 h = ((const float4*)(H + (size_t)s * dout))[c];
    float* yp = Y + (size_t)d * dout + (size_t)c * 4;
    atomicAdd(yp + 0, h.x * n);
    atomicAdd(yp + 1, h.y * n);
    atomicAdd(yp + 2, h.z * n);
    atomicAdd(yp + 3, h.w * n);
}

__global__ void k_bias(float* __restrict__ Y, const float* __restrict__ b,
                       int N, int dout, int relu) {
    long long t = (long long)blockIdx.x * blockDim.x + threadIdx.x;
    if (t >= (long long)N * dout) return;
    int f = (int)(t % dout);
    float v = Y[t] + b[f];
    if (relu) v = fmaxf(v, 0.0f);
    Y[t] = v;
}

// ---------------------------------------------------------------------------
// Launcher
// ---------------------------------------------------------------------------
extern "C" void kernel_launch(void* const* d_in, const int* in_sizes, int n_in,
                              void* d_out, int out_size, void* d_ws, size_t ws_size,
                              hipStream_t stream) {
    const float*     x  = (const float*)d_in[0];
    const long long* ei = (const long long*)d_in[1];   // int64 per reference
    const float*     ew = (const float*)d_in[2];
    const float*     W0 = (const float*)d_in[3];
    const float*     b0 = (const float*)d_in[4];
    const float*     W1 = (const float*)d_in[5];
    const float*     b1 = (const float*)d_in[6];
    const float*     W2 = (const float*)d_in[7];
    const float*     b2 = (const float*)d_in[8];

    const int DIN = 128, L0 = 128, L1 = 128, L2 = 64;
    const int N = in_sizes[0] / DIN;
    const int E = in_sizes[2];

    // Workspace carve-out (4 KB aligned blocks)
    char* wsp = (char*)d_ws;
    auto carve = [&](size_t bytes) -> char* {
        char* p = wsp;
        wsp += (bytes + 4095) & ~(size_t)4095;
        return p;
    };
    float* dinv  = (float*)carve((size_t)N * 4);          // deg -> dinv in place
    float* norm  = (float*)carve((size_t)E * 4);
    int*   src32 = (int*)  carve((size_t)E * 4);
    int*   dst32 = (int*)  carve((size_t)E * 4);
    float* H     = (float*)carve((size_t)N * 128 * 4);    // dense transform out
    float* Y     = (float*)carve((size_t)N * 128 * 4);    // aggregation buffer

    const int T = 256;
    k_deg_init   <<<(N + T - 1) / T, T, 0, stream>>>(dinv, N);
    k_deg_scatter<<<(E + T - 1) / T, T, 0, stream>>>(ei, ew, dinv, E);
    k_dinv       <<<(N + T - 1) / T, T, 0, stream>>>(dinv, N);
    k_norm       <<<(E + T - 1) / T, T, 0, stream>>>(ei, ew, dinv, norm, src32, dst32, E);

    auto layer = [&](const float* Xin, const float* Wt, const float* bt,
                     float* out, int din, int dout, int relu) {
        int tiles = ((N + 15) / 16) * (dout / 16);
        k_gemm_wmma<<<(tiles + 7) / 8, 256, 0, stream>>>(Xin, Wt, H, N, din, dout, tiles);
        long long tS = (long long)N * (dout / 4);
        k_selfloop<<<(unsigned)((tS + T - 1) / T), T, 0, stream>>>(H, dinv, out, N, dout);
        long long tE = (long long)E * (dout / 4);
        k_scatter <<<(unsigned)((tE + T - 1) / T), T, 0, stream>>>(H, src32, dst32, norm, out, E, dout);
        long long tB = (long long)N * dout;
        k_bias    <<<(unsigned)((tB + T - 1) / T), T, 0, stream>>>(out, bt, N, dout, relu);
    };

    layer(x, W0, b0, Y, DIN, L0, 1);                 // layer 0: x  -> Y (ReLU)
    layer(Y, W1, b1, Y, L0, L1, 1);                  // layer 1: Y  -> Y (ReLU)
    layer(Y, W2, b2, (float*)d_out, L1, L2, 0);      // layer 2: Y  -> out
}